// VQVAE_9844065042987
// MI455X (gfx1250) — compile-verified
//
#include <hip/hip_runtime.h>
#include <stdint.h>

// CDNA5 / gfx1250 — wave32. WMMA f32 16x16x4 for the VQ distance matmul,
// TDM (tensor_load_to_lds) to stage the full codebook in LDS per workgroup.
typedef __attribute__((ext_vector_type(2))) float v2f;
typedef __attribute__((ext_vector_type(8))) float v8f;
typedef __attribute__((ext_vector_type(4))) unsigned int u32x4;
typedef __attribute__((ext_vector_type(8))) int i32x8;
typedef __attribute__((ext_vector_type(4))) int i32x4;

#define Bk 4
#define Ck 4
#define Nk 4096            // H*W
#define Kk 8192            // codebook entries
#define Pk 16384           // B * N positions
#define OUT_LCB 65536
#define OUT_LCM 65537
#define OUT_IDX 65538

// Zero the two loss accumulators (harness poisons d_out once, never re-poisons).
__global__ __launch_bounds__(32) void vq_zero(float* __restrict__ lcb,
                                              float* __restrict__ lcm) {
    if (threadIdx.x == 0) { *lcb = 0.f; *lcm = 0.f; }
}

// One wave per 16-position tile; 8 waves (one block) per WGP.
// dist(m,n) = ||c_n||^2 - 2*q_m.c_n (+const per row) computed as
// D = A*B + C with A = -2*q (16x4 f32), B = codebook tile (4x16),
// C = broadcast ||c||^2 per column. Running argmin over 512 WMMA tiles,
// all operands LDS-resident after a single TDM copy of the codebook.
__global__ __launch_bounds__(256) void vq_main(const float* __restrict__ x,
                                               const float* __restrict__ pw,
                                               const float* __restrict__ pb,
                                               const float* __restrict__ cb,
                                               const float* __restrict__ qw,
                                               const float* __restrict__ qb,
                                               float* __restrict__ dec,
                                               float* lcb, float* lcm,
                                               float* __restrict__ idxo) {
    __shared__ __align__(16) float scb[Kk * Ck];   // 128 KB codebook
    __shared__ float scn[Kk];                      // 32 KB ||c||^2
    __shared__ float qsh[8][16][4];                // per-wave -2*prequant tile
    __shared__ int   ish[8][16];                   // winning indices

    const int wid  = threadIdx.x >> 5;
    const int lane = threadIdx.x & 31;
    const int m16  = lane & 15;       // column n (B/C/D) / row m (A)
    const int hh   = lane >> 4;       // lane half
    const int tile = blockIdx.x * 8 + wid;
    const int p0   = tile * 16;

    // ---- wave 0: kick TDM copy of the whole codebook into LDS ----
    if (wid == 0) {
        // D# group 0: count=1 | lds_addr | global_addr | type=2 ("image")
        uint64_t ga = (uint64_t)(uintptr_t)cb;        // global byte address
        uint32_t la = (uint32_t)(uintptr_t)scb;       // low 32 bits = LDS offset
        u32x4 g0;
        g0[0] = 1u;                                    // count=1, user desc
        g0[1] = la;                                    // lds_addr
        g0[2] = (uint32_t)ga;                          // global_addr[31:0]
        g0[3] = (uint32_t)((ga >> 32) & 0x1FFFFFFu)    // global_addr[56:32]
              | (2u << 30);                            // type=2
        // D# group 1: 1-D tensor, data_size=4B, dim0 = tile_dim0 = 32768
        i32x8 g1;
        g1[0] = (int)(2u << 16);                       // data_size=2 (4 bytes)
        g1[1] = (int)(32768u << 16);                   // tensor_dim0[15:0]
        g1[2] = (int)(1u << 16);                       // dim0 hi=0 | tensor_dim1=1
        g1[3] = (int)(32768u << 16);                   // tensor_dim1 hi=0 | tile_dim0
        g1[4] = 1;                                     // tile_dim1=1, tile_dim2=0
        g1[5] = 32768;                                 // tensor_dim0_stride lo
        g1[6] = (int)(32768u << 16);                   // stride0 hi=0 | stride1 lo
        g1[7] = 0;                                     // stride1 hi
        i32x4 z4 = {0, 0, 0, 0};
        i32x8 z8 = {0, 0, 0, 0, 0, 0, 0, 0};
        __builtin_amdgcn_tensor_load_to_lds(g0, g1, z4, z4, z8, 0);
    }

    // ---- all waves: codebook norms into LDS (overlaps the TDM) ----
    for (int r = threadIdx.x; r < Kk; r += 256) {
        const float4 row = ((const float4*)cb)[r];     // global_load_b128
        scn[r] = row.x * row.x + row.y * row.y + row.z * row.z + row.w * row.w;
    }

    // ---- prequant 4x4 for this wave's 16 positions (lanes 0..15) ----
    if (hh == 0) {
        int p = p0 + m16;
        int b = p >> 12;              // / N
        int n = p & (Nk - 1);
        const float* xb = x + (size_t)b * Ck * Nk + n;
        float xc[Ck];
#pragma unroll
        for (int c = 0; c < Ck; ++c) xc[c] = xb[c * Nk];
#pragma unroll
        for (int o = 0; o < Ck; ++o) {
            float q = pb[o];
#pragma unroll
            for (int c = 0; c < Ck; ++c) q += pw[o * Ck + c] * xc[c];
            qsh[wid][m16][o] = -2.0f * q;    // pre-scale: dist = cnorm - 2*dot
        }
    }

    if (wid == 0) __builtin_amdgcn_s_wait_tensorcnt(0);   // codebook landed
    __syncthreads();

    // ---- A operand, 16x4 f32 layout: VGPR j, half h -> K = j + 2h ----
    v2f a;
    a.x = qsh[wid][m16][2 * hh + 0];
    a.y = qsh[wid][m16][2 * hh + 1];

    float best[8];
    int   bidx[8];
#pragma unroll
    for (int v = 0; v < 8; ++v) { best[v] = 3.4e38f; bidx[v] = 0; }

#pragma unroll 4
    for (int kb = 0; kb < Kk; kb += 16) {
        int row = kb + m16;
        // B operand: channels (2h, 2h+1) of row kb+n -> one ds_load_b64
        v2f bop = *(const v2f*)&scb[row * Ck + 2 * hh];
        float cn = scn[row];                               // 16-lane broadcast
        v8f cacc = {cn, cn, cn, cn, cn, cn, cn, cn};

        v8f d = __builtin_amdgcn_wmma_f32_16x16x4_f32(
            false, a, false, bop, (short)0, cacc, false, false);

#pragma unroll
        for (int v = 0; v < 8; ++v) {
            float dv = d[v];
            if (dv < best[v]) { best[v] = dv; bidx[v] = row; }
        }
    }

    // ---- cross-lane argmin within each 16-lane half (columns 0..15) ----
#pragma unroll
    for (int v = 0; v < 8; ++v) {
#pragma unroll
        for (int off = 8; off >= 1; off >>= 1) {
            float ov = __shfl_xor(best[v], off, 32);
            int   oi = __shfl_xor(bidx[v], off, 32);
            if (ov < best[v] || (ov == best[v] && oi < bidx[v])) {
                best[v] = ov; bidx[v] = oi;
            }
        }
    }
    if (m16 == 0) {
#pragma unroll
        for (int v = 0; v < 8; ++v) ish[wid][v + 8 * hh] = bidx[v];
    }
    __syncthreads();

    // ---- gather + losses + postquant 4x4 (lanes 0..15, one position each) ----
    float partial = 0.f;
    if (hh == 0) {
        int p = p0 + m16;
        int b = p >> 12;
        int n = p & (Nk - 1);
        int kidx = ish[wid][m16];
        idxo[p] = (float)kidx;

        float qv[Ck], xc[Ck];
        const float* xb = x + (size_t)b * Ck * Nk + n;
#pragma unroll
        for (int c = 0; c < Ck; ++c) {
            qv[c] = scb[kidx * Ck + c];     // gather from LDS copy
            xc[c] = xb[c * Nk];
            float dd = qv[c] - xc[c];
            partial += dd * dd;
        }
        float* db = dec + (size_t)b * Ck * Nk + n;
#pragma unroll
        for (int o = 0; o < Ck; ++o) {
            float s = qb[o];
#pragma unroll
            for (int c = 0; c < Ck; ++c) s += qw[o * Ck + c] * qv[c];
            db[o * Nk] = s;
        }
    }
    // wave-level loss reduction; both losses share the same forward value
#pragma unroll
    for (int off = 16; off >= 1; off >>= 1) partial += __shfl_xor(partial, off, 32);
    if (lane == 0) {
        float v = partial * (1.0f / (float)(Pk * Ck));
        atomicAdd(lcb, v);
        atomicAdd(lcm, v);
    }
}

extern "C" void kernel_launch(void* const* d_in, const int* in_sizes, int n_in,
                              void* d_out, int out_size, void* d_ws, size_t ws_size,
                              hipStream_t stream) {
    const float* x   = (const float*)d_in[0];
    const float* pw  = (const float*)d_in[1];
    const float* pb  = (const float*)d_in[2];
    const float* cbk = (const float*)d_in[3];
    const float* qw  = (const float*)d_in[4];
    const float* qb  = (const float*)d_in[5];

    float* out  = (float*)d_out;
    float* dec  = out;                 // [B,C,H,W] = 65536 floats
    float* lcb  = out + OUT_LCB;       // codebook_loss
    float* lcm  = out + OUT_LCM;       // commitment_loss
    float* idxo = out + OUT_IDX;       // [B,H,W] = 16384 (as float)
    (void)d_ws; (void)ws_size; (void)in_sizes; (void)n_in; (void)out_size;

    vq_zero<<<1, 32, 0, stream>>>(lcb, lcm);
    vq_main<<<Pk / 16 / 8, 256, 0, stream>>>(x, pw, pb, cbk, qw, qb,
                                             dec, lcb, lcm, idxo);
}